// gplayer_50148038148313
// MI455X (gfx1250) — compile-verified
//
#include <hip/hip_runtime.h>
#include <math.h>

// ---------------------------------------------------------------------------
// Multi-fidelity GP log-marginal-likelihood on MI455X (gfx1250).
//   K build  : bandwidth-trivial (105 MB @ 23.3 TB/s)
//   Cholesky : n^3/3 ~= 45 GFLOP, SYRK trailing update runs on
//              V_WMMA_F32_16X16X4_F32 (native fp32 matrix pipe -> exact fp32)
//   solve    : quad = ||L^-1 y||^2, logdet = 2*sum(log diag L)
// ---------------------------------------------------------------------------

typedef float v2f __attribute__((ext_vector_type(2)));
typedef float v8f __attribute__((ext_vector_type(8)));

#define NB     128          // Cholesky panel / tile size
#define PD     129          // LDS pitch for 128-wide tiles (bank-conflict free)
#define KC     32           // K-chunk staged in LDS for SYRK
#define KP     34           // LDS pitch for K-chunks (34*r mod 64 distinct)

#define WMMA_F32(accv, av, bv)                                                 \
  accv = __builtin_amdgcn_wmma_f32_16x16x4_f32(false, av, false, bv,           \
                                               (short)0, accv, false, false)

#define LOG_2PI 1.8378770664093453f

// ---------------------------------------------------------------------------
// Kernel 1: assemble K (row-major n x n) from x / hyp.  n_low read on device.
// ---------------------------------------------------------------------------
__global__ __launch_bounds__(256) void build_K(float* __restrict__ K,
                                               const float* __restrict__ x,
                                               const float* __restrict__ hyp,
                                               const int* __restrict__ nlow_p,
                                               int n) {
  int j = blockIdx.x * 16 + (threadIdx.x & 15);
  int i = blockIdx.y * 16 + (threadIdx.x >> 4);
  if (i >= n || j >= n) return;
  const int nlow = *nlow_p;
  const float h0 = hyp[0], h1 = hyp[1], h2 = hyp[2];
  const float h3 = hyp[3], h4 = hyp[4], h5 = hyp[5], rho = hyp[6];
  const float d  = x[i] - x[j];
  const float d2 = d * d;
  const bool hi_i = (i >= nlow), hi_j = (j >= nlow);
  float v = h0 * h0 * __expf(-h1 * h1 * d2);
  if (hi_i) v *= rho;
  if (hi_j) v *= rho;
  if (hi_i && hi_j) v += h3 * h3 * __expf(-h4 * h4 * d2);
  if (i == j) v += hi_i ? h5 * h5 : h2 * h2;
  K[(size_t)i * n + j] = v;
}

// ---------------------------------------------------------------------------
// Kernel 2: factor the 128x128 diagonal block (in LDS), write L11 back.
// ---------------------------------------------------------------------------
__global__ __launch_bounds__(256) void potrf_diag(float* __restrict__ K,
                                                  int n, int kblk) {
  extern __shared__ float As[];                 // [NB][PD]
  const int tid = threadIdx.x;
  const int g0  = kblk * NB;
  const int mb  = min(NB, n - g0);

  for (int p = tid; p < mb * NB; p += 256) {
    int r = p >> 7, c = p & 127;
    if (c < mb) As[r * PD + c] = K[(size_t)(g0 + r) * n + (g0 + c)];
  }
  __syncthreads();

  for (int j = 0; j < mb; ++j) {
    if (tid == 0) As[j * PD + j] = sqrtf(As[j * PD + j]);
    __syncthreads();
    const float dj = As[j * PD + j];
    for (int i = j + 1 + tid; i < mb; i += 256) As[i * PD + j] /= dj;
    __syncthreads();
    const int m = mb - 1 - j;                   // trailing square (symmetric)
    for (int p = tid; p < m * m; p += 256) {
      int r = j + 1 + p / m, c = j + 1 + p % m;
      As[r * PD + c] -= As[r * PD + j] * As[c * PD + j];
    }
    __syncthreads();
  }

  for (int p = tid; p < mb * NB; p += 256) {
    int r = p >> 7, c = p & 127;
    if (c <= r) K[(size_t)(g0 + r) * n + (g0 + c)] = As[r * PD + c];
  }
}

// ---------------------------------------------------------------------------
// Kernel 3: panel TRSM  L21 * L11^T = A21.  One 128-row block per workgroup;
// each thread forward-substitutes its own row (no inner barriers).
// ---------------------------------------------------------------------------
__global__ __launch_bounds__(128) void trsm_panel(float* __restrict__ K,
                                                  int n, int kblk) {
  extern __shared__ float sm[];
  float* Ls = sm;                               // [NB][PD]  L11
  float* Xs = sm + NB * PD;                     // [NB][PD]  row block
  const int tid = threadIdx.x;
  const int k0  = kblk * NB;
  const int r0  = (kblk + 1 + blockIdx.x) * NB;
  const int mb  = min(NB, n - r0);

  for (int p = tid; p < NB * NB; p += 128) {
    int r = p >> 7, c = p & 127;
    Ls[r * PD + c] = K[(size_t)(k0 + r) * n + (k0 + c)];
    Xs[r * PD + c] = (r < mb) ? K[(size_t)(r0 + r) * n + (k0 + c)] : 0.0f;
  }
  __syncthreads();

  if (tid < mb) {
    float* xrow = &Xs[tid * PD];
    for (int j = 0; j < NB; ++j) {
      float s = xrow[j];
      for (int k = 0; k < j; ++k) s -= xrow[k] * Ls[j * PD + k];
      xrow[j] = s / Ls[j * PD + j];
    }
  }
  __syncthreads();

  for (int p = tid; p < NB * NB; p += 128) {
    int r = p >> 7, c = p & 127;
    if (r < mb) K[(size_t)(r0 + r) * n + (k0 + c)] = Xs[r * PD + c];
  }
}

__device__ inline void store_tile_sub(float* __restrict__ K, int n,
                                      int row0, int col, v8f acc) {
  if (col >= n) return;
#pragma unroll
  for (int r = 0; r < 8; ++r) {
    int rr = row0 + r;
    if (rr < n) K[(size_t)rr * n + col] -= acc[r];
  }
}

// ---------------------------------------------------------------------------
// Kernel 4: SYRK trailing update A[bi,bj] -= L[bi,k] * L[bj,k]^T  via
// V_WMMA_F32_16X16X4_F32.  128x128 tile / workgroup, 8 waves, wave = 32x64.
// ---------------------------------------------------------------------------
__global__ __launch_bounds__(256) void syrk_update(float* __restrict__ K,
                                                   int n, int kblk) {
  const int bi = kblk + 1 + blockIdx.y;
  const int bj = kblk + 1 + blockIdx.x;
  if (bj > bi) return;                          // lower trailing part only

  extern __shared__ float sm[];
  float* Li = sm;                               // [NB][KP]
  float* Lj = sm + NB * KP;                     // [NB][KP]

  const int tid  = threadIdx.x;
  const int lane = tid & 31;
  const int wave = tid >> 5;                    // 0..7
  const int wm   = (wave >> 1) * 32;            // wave M offset: 0,32,64,96
  const int wn   = (wave & 1) * 64;             // wave N offset: 0,64
  const int l16  = lane & 15;
  const int hi   = lane >> 4;                   // 0: K pair {0,1}, 1: {2,3}

  const int row0 = bi * NB, col0 = bj * NB, kk0 = kblk * NB;

  v8f a00 = {}, a01 = {}, a02 = {}, a03 = {};
  v8f a10 = {}, a11 = {}, a12 = {}, a13 = {};

  for (int kc = 0; kc < NB; kc += KC) {
    // stage 128x32 chunks of both panels into LDS (coalesced, zero-fill OOB)
    for (int p = tid; p < NB * KC; p += 256) {
      int r = p >> 5, c = p & 31;
      int gk = kk0 + kc + c;
      Li[r * KP + c] = (row0 + r < n) ? K[(size_t)(row0 + r) * n + gk] : 0.0f;
      Lj[r * KP + c] = (col0 + r < n) ? K[(size_t)(col0 + r) * n + gk] : 0.0f;
    }
    __syncthreads();

#pragma unroll
    for (int ks = 0; ks < KC; ks += 4) {
      const int cb = ks + (hi << 1);
      // A fragments (16x4 f32): lanes 0-15 -> K={cb,cb+1} of rows M=l16
      v2f fa0 = { Li[(wm + l16) * KP + cb],      Li[(wm + l16) * KP + cb + 1] };
      v2f fa1 = { Li[(wm + 16 + l16) * KP + cb], Li[(wm + 16 + l16) * KP + cb + 1] };
      // B fragments (4x16 f32) of Lj^T: B[k][nn] = Lj[nn][k]
      v2f fb0 = { Lj[(wn + l16) * KP + cb],      Lj[(wn + l16) * KP + cb + 1] };
      v2f fb1 = { Lj[(wn + 16 + l16) * KP + cb], Lj[(wn + 16 + l16) * KP + cb + 1] };
      v2f fb2 = { Lj[(wn + 32 + l16) * KP + cb], Lj[(wn + 32 + l16) * KP + cb + 1] };
      v2f fb3 = { Lj[(wn + 48 + l16) * KP + cb], Lj[(wn + 48 + l16) * KP + cb + 1] };

      WMMA_F32(a00, fa0, fb0); WMMA_F32(a01, fa0, fb1);
      WMMA_F32(a02, fa0, fb2); WMMA_F32(a03, fa0, fb3);
      WMMA_F32(a10, fa1, fb0); WMMA_F32(a11, fa1, fb1);
      WMMA_F32(a12, fa1, fb2); WMMA_F32(a13, fa1, fb3);
    }
    __syncthreads();
  }

  // write back: C/D layout -> VGPR r holds (M = r + 8*hi, N = l16)
  const int rb0 = row0 + wm + (hi << 3);        // M sub-tile 0
  const int rb1 = rb0 + 16;                     // M sub-tile 1
  const int cb0 = col0 + wn + l16;
  store_tile_sub(K, n, rb0, cb0 +  0, a00);
  store_tile_sub(K, n, rb0, cb0 + 16, a01);
  store_tile_sub(K, n, rb0, cb0 + 32, a02);
  store_tile_sub(K, n, rb0, cb0 + 48, a03);
  store_tile_sub(K, n, rb1, cb0 +  0, a10);
  store_tile_sub(K, n, rb1, cb0 + 16, a11);
  store_tile_sub(K, n, rb1, cb0 + 32, a12);
  store_tile_sub(K, n, rb1, cb0 + 48, a13);
}

// ---------------------------------------------------------------------------
// Kernel 5: forward solve L z = y (one workgroup), quad = sum z^2,
// logdet = 2*sum log Ljj;  out = quad + logdet + n*log(2*pi)
// ---------------------------------------------------------------------------
#define SOLVE_T 1024
#define MAXROWS 8
__global__ __launch_bounds__(SOLVE_T) void solve_logdet(
    const float* __restrict__ K, const float* __restrict__ y,
    float* __restrict__ out, int n) {
  extern __shared__ float sm[];
  float* zs = sm;                               // [n]
  float* rq = sm + n;                           // [SOLVE_T]
  float* rl = rq + SOLVE_T;                     // [SOLVE_T]
  const int tid = threadIdx.x;

  float acc[MAXROWS];
#pragma unroll
  for (int r = 0; r < MAXROWS; ++r) acc[r] = 0.0f;
  float logs = 0.0f;

  for (int j = 0; j < n; ++j) {
#pragma unroll
    for (int r = 0; r < MAXROWS; ++r) {
      int i = tid + (r << 10);
      if (i == j) {                             // this thread owns row j
        float dj = K[(size_t)j * n + j];
        zs[j] = (y[j] - acc[r]) / dj;
        logs += __logf(dj);
      }
    }
    __syncthreads();
    const float zj = zs[j];
#pragma unroll
    for (int r = 0; r < MAXROWS; ++r) {
      int i = tid + (r << 10);
      if (i > j && i < n) acc[r] += K[(size_t)i * n + j] * zj;
    }
  }

  float q = 0.0f;
  for (int i = tid; i < n; i += SOLVE_T) q += zs[i] * zs[i];
  rq[tid] = q;
  rl[tid] = logs;
  __syncthreads();
  for (int s = SOLVE_T / 2; s > 0; s >>= 1) {
    if (tid < s) { rq[tid] += rq[tid + s]; rl[tid] += rl[tid + s]; }
    __syncthreads();
  }
  if (tid == 0) out[0] = rq[0] + 2.0f * rl[0] + (float)n * LOG_2PI;
}

// ---------------------------------------------------------------------------
extern "C" void kernel_launch(void* const* d_in, const int* in_sizes, int n_in,
                              void* d_out, int out_size, void* d_ws,
                              size_t ws_size, hipStream_t stream) {
  const float* x    = (const float*)d_in[0];
  const float* y    = (const float*)d_in[1];
  const float* hyp  = (const float*)d_in[2];
  const int*   nlow = (const int*)d_in[4];
  const int n = in_sizes[0];

  float* K   = (float*)d_ws;                    // n*n f32
  float* out = (float*)d_out;

  const int POTRF_SMEM = NB * PD * sizeof(float);          // 66,048 B
  const int TRSM_SMEM  = 2 * NB * PD * sizeof(float);      // 132,096 B
  const int SYRK_SMEM  = 2 * NB * KP * sizeof(float);      // 34,816 B
  (void)hipFuncSetAttribute((const void*)potrf_diag,
      hipFuncAttributeMaxDynamicSharedMemorySize, POTRF_SMEM);
  (void)hipFuncSetAttribute((const void*)trsm_panel,
      hipFuncAttributeMaxDynamicSharedMemorySize, TRSM_SMEM);

  // 1) assemble K
  dim3 gb((n + 15) / 16, (n + 15) / 16);
  build_K<<<gb, 256, 0, stream>>>(K, x, hyp, nlow, n);

  // 2) blocked right-looking Cholesky
  const int nb = (n + NB - 1) / NB;
  for (int k = 0; k < nb; ++k) {
    potrf_diag<<<1, 256, POTRF_SMEM, stream>>>(K, n, k);
    const int m = nb - k - 1;
    if (m > 0) {
      trsm_panel<<<m, 128, TRSM_SMEM, stream>>>(K, n, k);
      syrk_update<<<dim3(m, m), 256, SYRK_SMEM, stream>>>(K, n, k);
    }
  }

  // 3) forward solve + logdet -> scalar
  const int SOLVE_SMEM = (n + 2 * SOLVE_T) * sizeof(float);
  solve_logdet<<<1, SOLVE_T, SOLVE_SMEM, stream>>>(K, y, out, n);
}